// MultiheadAttention_24653112279580
// MI455X (gfx1250) — compile-verified
//
#include <hip/hip_runtime.h>

// ---------------------------------------------------------------------------
// PDE-attention (Adams-Bashforth banded "attention") for MI455X / gfx1250.
//
// band_mask(S,4) zeroes everything except j in {i-3..i}, so the S x S
// attention collapses to a width-4 banded op (O(S*4*D), not O(S^2*D)).
// The v-transform is a circulant 3-pt / 2-pt channel stencil (D2/D1 masks
// of Wv2/Wv). Only the q/k projections are real GEMMs -> exact f32 via
// v_wmma_f32_16x16x4_f32 (inputs/outputs are f32; the sum-normalization
// divides by a small signed sum, so we keep full precision).
// ---------------------------------------------------------------------------

typedef __attribute__((ext_vector_type(2))) float v2f;
typedef __attribute__((ext_vector_type(8))) float v8f;

#define S_LEN   2048
#define BATCH   4
#define DM      256      // d_model
#define NH      4
#define DH      64
#define ROWS    (S_LEN * BATCH)   // 8192 flattened (s,b) rows

static constexpr float kScale = 0.125f;                   // 1/sqrt(64)
static constexpr float kCsc   = 40.74366543152521f;       // NX/(2*pi) = 256/(2*pi)

// ---------------------------------------------------------------------------
// Kernel 1: q = h @ Wq^T, k = h @ Wk^T  via V_WMMA_F32_16X16X4_F32.
//
// Each wave computes a 16x64 output stripe = 4 independent 16x16 tiles:
//   - 4 separate accumulators -> 4 interleaved WMMA chains (hides the
//     multi-cycle WMMA result latency; a single chain would serialize),
//   - A fragment reused across the 4 tiles -> 1 A-load + 4 B-loads per
//     4 WMMAs (1.25 loads/wmma instead of 2).
// Block = 8 waves: waves 0-3 cover q cols 0..255, waves 4-7 cover k.
//
// Layouts (CDNA5 ISA 7.12.2, 32-bit):
//   A 16x4 : lane l -> row m = l&15 ; lane-half selects K-pair {0,1}/{2,3};
//            VGPR0/VGPR1 = the two K values -> contiguous float2 per lane.
//   B 4x16 : mirrored, lane l -> col n = l&15 ; same K-pair split.
//   C/D    : VGPR v, lane l -> (m = v + 8*(l>=16), n = l&15).
// ---------------------------------------------------------------------------
__global__ __launch_bounds__(256)
void qk_gemm_wmma(const float* __restrict__ h,
                  const float* __restrict__ Wq,
                  const float* __restrict__ Wk,
                  float* __restrict__ qout,
                  float* __restrict__ kout)
{
    const int wave = threadIdx.x >> 5;
    const int lane = threadIdx.x & 31;
    const int lrow = lane & 15;           // A: row m   / B: col n
    const int lkp  = (lane >> 4) << 1;    // K-pair base: 0 or 2

    const int r0  = blockIdx.x * 16;      // output row tile (16 rows)
    const int mat = wave >> 2;            // 0 = q, 1 = k
    const int cg  = wave & 3;             // 64-col group within d_model
    const float* W   = mat ? Wk  : Wq;
    float*       out = mat ? kout : qout;

    // A[m][k] = h[r0+m, kk+k] ; B[k][n] = W[i0+n, kk+k]  (both row-major)
    const float* aptr = h + (r0 + lrow) * DM + lkp;
    const float* b0   = W + (cg * 64 + lrow) * DM + lkp;   // tile n-offset 0
    const float* b1   = b0 + 16 * DM;                      // +16 cols
    const float* b2   = b0 + 32 * DM;                      // +32 cols
    const float* b3   = b0 + 48 * DM;                      // +48 cols

    v8f acc0 = {}, acc1 = {}, acc2 = {}, acc3 = {};
#pragma unroll 4
    for (int kk = 0; kk < DM; kk += 4) {
        const v2f a   = *(const v2f*)(aptr + kk);
        const v2f bb0 = *(const v2f*)(b0 + kk);
        const v2f bb1 = *(const v2f*)(b1 + kk);
        const v2f bb2 = *(const v2f*)(b2 + kk);
        const v2f bb3 = *(const v2f*)(b3 + kk);
        // (neg_a, A, neg_b, B, c_mod, C, reuse_a, reuse_b)
        acc0 = __builtin_amdgcn_wmma_f32_16x16x4_f32(false, a, false, bb0,
                                                     (short)0, acc0, false, false);
        acc1 = __builtin_amdgcn_wmma_f32_16x16x4_f32(false, a, false, bb1,
                                                     (short)0, acc1, false, false);
        acc2 = __builtin_amdgcn_wmma_f32_16x16x4_f32(false, a, false, bb2,
                                                     (short)0, acc2, false, false);
        acc3 = __builtin_amdgcn_wmma_f32_16x16x4_f32(false, a, false, bb3,
                                                     (short)0, acc3, false, false);
    }

    // Store D tiles: lane l, VGPR v -> row (v + 8*(l>>4)), col (l&15)
    float* ob = out + (r0 + ((lane >> 4) * 8)) * DM + cg * 64 + lrow;
#pragma unroll
    for (int v = 0; v < 8; ++v) {
        ob[v * DM +  0] = acc0[v];
        ob[v * DM + 16] = acc1[v];
        ob[v * DM + 32] = acc2[v];
        ob[v * DM + 48] = acc3[v];
    }
}

// ---------------------------------------------------------------------------
// Kernel 2: banded AB attention + inline PDE value stencil.
// One wave32 per (i, b, head). Lane owns channels c = n*64 + 2*lane, c+1.
//   score_t = (q_i . k_{i-t}) * SCALE          (t = 0..min(i,3))
//   masked_t = K[i, i-t] * score_t
//   prob_t  = masked_t / (sum_t masked_t + 1e-6)
//   out_i   = sum_t prob_t * v_{i-t}
// v_j[c] = Csc*(h[c-1]*Wv2[c,c-1] + h[c]*Wv2[c,c] + h[c+1]*Wv2[c,c+1])
//        + h[c]*(h[c-1]*Wv[c,c-1] + h[c]*Wv[c,c])        (channels mod 256)
// ---------------------------------------------------------------------------
__global__ __launch_bounds__(256)
void band_attn(const float* __restrict__ h,
               const float* __restrict__ Wv,
               const float* __restrict__ Wv2,
               const float* __restrict__ qbuf,
               const float* __restrict__ kbuf,
               float* __restrict__ out)
{
    const int lane = threadIdx.x & 31;
    const int g = blockIdx.x * 8 + (threadIdx.x >> 5);   // wave id
    const int n = g & (NH - 1);
    const int b = (g >> 2) & (BATCH - 1);
    const int i = g >> 4;                                // 0..2047 (uniform per wave)

    const int c0 = n * DH + 2 * lane;                    // channel pair (c0, c0+1)
    const int c1 = c0 + 1;                               // <= 255, never wraps
    const int c0m = (c0 + DM - 1) & (DM - 1);            // circulant neighbors
    const int c1p = (c1 + 1) & (DM - 1);

    const v2f q2 = *(const v2f*)(qbuf + (i * BATCH + b) * DM + c0);

    // Stencil weights (t-invariant, hoisted out of the band loop)
    const float w2_0m = Wv2[c0 * DM + c0m], w2_00 = Wv2[c0 * DM + c0],
                w2_0p = Wv2[c0 * DM + c1];
    const float w2_1m = Wv2[c1 * DM + c0],  w2_10 = Wv2[c1 * DM + c1],
                w2_1p = Wv2[c1 * DM + c1p];
    const float w1_0m = Wv [c0 * DM + c0m], w1_00 = Wv [c0 * DM + c0];
    const float w1_1m = Wv [c1 * DM + c0],  w1_10 = Wv [c1 * DM + c1];

    // Adams-Bashforth band coefficients K[i, i-t] (startup rows use low order)
    const int T = (i >= 3) ? 4 : (i + 1);
    float coef[4];
    if (i >= 3)      { coef[0] =  55.f/24.f; coef[1] = -59.f/24.f;
                       coef[2] =  37.f/24.f; coef[3] =  -9.f/24.f; }
    else if (i == 2) { coef[0] =  23.f/12.f; coef[1] = -16.f/12.f;
                       coef[2] =   5.f/12.f; coef[3] = 0.f; }
    else if (i == 1) { coef[0] = 1.5f; coef[1] = -0.5f; coef[2] = 0.f; coef[3] = 0.f; }
    else             { coef[0] = 1.0f; coef[1] = 0.f;  coef[2] = 0.f; coef[3] = 0.f; }

    float masked[4], vx[4], vy[4];
    float denom = 1e-6f;
    for (int t = 0; t < T; ++t) {
        const int j  = i - t;
        const int rb = (j * BATCH + b) * DM;

        // banded score: 64-wide dot, 2 elements/lane + wave32 butterfly reduce
        const v2f k2 = *(const v2f*)(kbuf + rb + c0);
        float p = q2.x * k2.x + q2.y * k2.y;
        for (int off = 16; off > 0; off >>= 1)
            p += __shfl_xor(p, off, 32);
        masked[t] = coef[t] * (p * kScale);
        denom += masked[t];

        // inline v_j for channels c0, c1 from the h stencil
        const float hm  = h[rb + c0m];
        const float h0  = h[rb + c0];
        const float hp  = h[rb + c1];
        const float hpp = h[rb + c1p];
        vx[t] = kCsc * (hm * w2_0m + h0 * w2_00 + hp * w2_0p)
              + h0   * (hm * w1_0m + h0 * w1_00);
        vy[t] = kCsc * (h0 * w2_1m + hp * w2_10 + hpp * w2_1p)
              + hp   * (h0 * w1_1m + hp * w1_10);
    }

    float ox = 0.f, oy = 0.f;
    for (int t = 0; t < T; ++t) {
        const float pr = masked[t] / denom;
        ox += pr * vx[t];
        oy += pr * vy[t];
    }
    v2f o; o.x = ox; o.y = oy;
    *(v2f*)(out + (i * BATCH + b) * DM + c0) = o;
}

// ---------------------------------------------------------------------------
extern "C" void kernel_launch(void* const* d_in, const int* in_sizes, int n_in,
                              void* d_out, int out_size, void* d_ws, size_t ws_size,
                              hipStream_t stream)
{
    (void)in_sizes; (void)n_in; (void)out_size; (void)ws_size;
    const float* h   = (const float*)d_in[0];
    const float* Wq  = (const float*)d_in[1];
    const float* Wk  = (const float*)d_in[2];
    const float* Wv  = (const float*)d_in[3];
    const float* Wv2 = (const float*)d_in[4];

    float* qbuf = (float*)d_ws;                       //  8 MB
    float* kbuf = qbuf + (size_t)ROWS * DM;           // +8 MB
    float* outp = (float*)d_out;

    // q/k GEMMs: 512 blocks, one 16-row stripe each; 8 waves/block
    // (4 waves x 64 cols for q, 4 for k), 4 WMMA chains per wave.
    qk_gemm_wmma<<<dim3(ROWS / 16), dim3(256), 0, stream>>>(h, Wq, Wk, qbuf, kbuf);

    // Banded attention: 32768 waves, 8 per block.
    band_attn<<<dim3((S_LEN * BATCH * NH) / 8), dim3(256), 0, stream>>>(
        h, Wv, Wv2, qbuf, kbuf, outp);
}